// MultinomialResampler_84327387890378
// MI455X (gfx1250) — compile-verified
//
#include <hip/hip_runtime.h>
#include <stdint.h>

#define BATCH  16
#define NPART  65536
#define DIM    64
#define TILE   4096
#define NTILES (NPART / TILE)   // 16

typedef __attribute__((ext_vector_type(2))) float v2f;
typedef __attribute__((ext_vector_type(8))) float v8f;

// ---------------------------------------------------------------- K1: per-batch max
__global__ __launch_bounds__(1024) void k_max(const float* __restrict__ w,
                                              float* __restrict__ d_max) {
    const int b = blockIdx.x;
    const float* wb = w + (size_t)b * NPART;
    float m = -3.402823466e38f;
    for (int j = threadIdx.x; j < NPART; j += 1024) m = fmaxf(m, wb[j]);
    __shared__ float s[1024];
    s[threadIdx.x] = m;
    __syncthreads();
    for (int off = 512; off > 0; off >>= 1) {
        if (threadIdx.x < off) s[threadIdx.x] = fmaxf(s[threadIdx.x], s[threadIdx.x + off]);
        __syncthreads();
    }
    if (threadIdx.x == 0) d_max[b] = s[0];
}

// ---------------------------------------------------------------- K2: tile sums via WMMA
// Each wave accumulates 8 x 64 = 512 exp() values through v_wmma_f32_16x16x4_f32 with
// B = ones: D[m][n] = sum_k A[m][k] + C. Summing the 8 accumulator VGPRs gives the
// half-wave partial; shfl_xor(16) folds the two halves.
__global__ __launch_bounds__(256) void k_tilesum(const float* __restrict__ w,
                                                 const float* __restrict__ d_max,
                                                 float* __restrict__ d_tileSum) {
    const int blk  = blockIdx.x;            // b * NTILES + tile
    const int b    = blk / NTILES;
    const int t    = blk % NTILES;
    const float mx = d_max[b];
    const int wave = threadIdx.x >> 5;      // 0..7
    const int lane = threadIdx.x & 31;

    const float* base = w + (size_t)b * NPART + (size_t)t * TILE + wave * 512 + lane * 2;

    v8f c = {};
    v2f ones; ones.x = 1.0f; ones.y = 1.0f;
    #pragma unroll
    for (int i = 0; i < 8; ++i) {
        v2f a;
        a.x = __expf(base[i * 64 + 0] - mx);
        a.y = __expf(base[i * 64 + 1] - mx);
        c = __builtin_amdgcn_wmma_f32_16x16x4_f32(false, a, false, ones,
                                                  (short)0, c, false, false);
    }
    float partial = c[0] + c[1] + c[2] + c[3] + c[4] + c[5] + c[6] + c[7];
    partial += __shfl_xor(partial, 16, 32);   // S_low + S_high -> wave total (all lanes)

    __shared__ float s[8];
    if (lane == 0) s[wave] = partial;
    __syncthreads();
    if (threadIdx.x == 0) {
        float tot = 0.0f;
        #pragma unroll
        for (int i = 0; i < 8; ++i) tot += s[i];
        d_tileSum[blk] = tot;
    }
}

// ---------------------------------------------------------------- K3: scan tile sums (tiny)
__global__ void k_scan_tiles(const float* __restrict__ d_tileSum,
                             float* __restrict__ d_tileOff,
                             float* __restrict__ d_tot) {
    const int b = threadIdx.x;
    if (b < BATCH) {
        float run = 0.0f;
        for (int t = 0; t < NTILES; ++t) {
            d_tileOff[b * NTILES + t] = run;
            run += d_tileSum[b * NTILES + t];
        }
        d_tot[b] = run;
    }
}

// ---------------------------------------------------------------- K4: inclusive CDF
__global__ __launch_bounds__(1024) void k_cdf(const float* __restrict__ w,
                                              const float* __restrict__ d_max,
                                              const float* __restrict__ d_tileOff,
                                              float* __restrict__ cdf) {
    const int blk  = blockIdx.x;            // b * NTILES + tile
    const int b    = blk / NTILES;
    const int t    = blk % NTILES;
    const float mx = d_max[b];
    const int tid  = threadIdx.x;           // 0..1023, 4 contiguous elems each

    const size_t base = (size_t)b * NPART + (size_t)t * TILE;
    const float4 x = ((const float4*)(w + base))[tid];
    const float p0 = __expf(x.x - mx);
    const float p1 = __expf(x.y - mx);
    const float p2 = __expf(x.z - mx);
    const float p3 = __expf(x.w - mx);
    const float q0 = p0, q1 = q0 + p1, q2 = q1 + p2, q3 = q2 + p3;

    __shared__ float s[1024];
    s[tid] = q3;
    __syncthreads();
    for (int off = 1; off < 1024; off <<= 1) {       // Hillis-Steele inclusive scan
        const float v = (tid >= off) ? s[tid - off] : 0.0f;
        __syncthreads();
        s[tid] += v;
        __syncthreads();
    }
    const float excl = s[tid] - q3 + d_tileOff[blk];
    float4 o;
    o.x = excl + q0; o.y = excl + q1; o.z = excl + q2; o.w = excl + q3;
    ((float4*)(cdf + base))[tid] = o;
}

// ---------------------------------------------------------------- K5: sample indices
__global__ __launch_bounds__(256) void k_sample(const float* __restrict__ cdf,
                                                const float* __restrict__ d_tot,
                                                uint32_t* __restrict__ idx,
                                                float* __restrict__ out_w) {
    const int r = blockIdx.x * 256 + threadIdx.x;    // 0 .. B*N-1
    const int b = r >> 16;                           // N == 65536

    uint32_t h = (uint32_t)r ^ 0x9E3779B9u;          // deterministic hash-based uniform
    h ^= h >> 16; h *= 0x7feb352du;
    h ^= h >> 15; h *= 0x846ca68bu;
    h ^= h >> 16;
    const float u = (float)(h >> 8) * (1.0f / 16777216.0f);
    const float target = u * d_tot[b];

    const float* c = cdf + ((size_t)b << 16);
    int lo = 0, hi = NPART;                          // first j with c[j] > target
    #pragma unroll
    for (int step = 0; step < 16; ++step) {
        const int mid = (lo + hi) >> 1;
        if (c[mid] <= target) lo = mid + 1; else hi = mid;
    }
    const int j = (lo < NPART) ? lo : (NPART - 1);
    idx[r]   = (uint32_t)j;
    out_w[r] = -11.090354888959125f;                 // -log(65536)
}

// ---------------------------------------------------------------- K6: gather via async LDS DMA
// Each thread moves one float4 of a gathered row: global -> LDS (async DMA),
// wait asynccnt, LDS -> global (async DMA). Data never touches VGPRs.
__global__ __launch_bounds__(256) void k_gather(const float* __restrict__ state,
                                                const uint32_t* __restrict__ idx,
                                                float* __restrict__ out) {
    __shared__ float4 smem[256];

    const int g = blockIdx.x * 256 + threadIdx.x;    // float4 slot: 0 .. B*N*16-1
    const int r = g >> 4;                            // sample row
    const int c = g & 15;                            // float4 within row
    const int b = r >> 16;                           // N == 65536

    const uint32_t j = idx[r];
    const float* src = state + ((size_t)b * NPART + j) * DIM + (size_t)c * 4;
    float*       dst = out   + (size_t)r * DIM + (size_t)c * 4;

    // low 32 bits of a generic LDS pointer == LDS byte address
    const uint32_t lds = (uint32_t)(uintptr_t)(&smem[threadIdx.x]);

    asm volatile("global_load_async_to_lds_b128 %0, %1, off"
                 :: "v"(lds), "v"(src) : "memory");
    asm volatile("s_wait_asynccnt 0x0" ::: "memory");
    asm volatile("global_store_async_from_lds_b128 %0, %1, off"
                 :: "v"(dst), "v"(lds) : "memory");
    asm volatile("s_wait_asynccnt 0x0" ::: "memory");
}

// ---------------------------------------------------------------- launch
extern "C" void kernel_launch(void* const* d_in, const int* in_sizes, int n_in,
                              void* d_out, int out_size, void* d_ws, size_t ws_size,
                              hipStream_t stream) {
    const float* state  = (const float*)d_in[0];     // [16, 65536, 64] f32
    const float* weight = (const float*)d_in[1];     // [16, 65536] f32
    float* out_state = (float*)d_out;                                // [16,65536,64]
    float* out_w     = (float*)d_out + (size_t)BATCH * NPART * DIM;  // [16,65536]

    float* ws        = (float*)d_ws;
    float* d_max     = ws;            // 16
    float* d_tot     = ws + 16;       // 16
    float* d_tileSum = ws + 32;       // 256
    float* d_tileOff = ws + 288;      // 256
    float* cdf       = ws + 1024;                                 // B*N floats (4 MiB)
    uint32_t* idx    = (uint32_t*)(ws + 1024 + (size_t)BATCH * NPART); // B*N u32 (4 MiB)

    k_max       <<<BATCH,             1024, 0, stream>>>(weight, d_max);
    k_tilesum   <<<BATCH * NTILES,     256, 0, stream>>>(weight, d_max, d_tileSum);
    k_scan_tiles<<<1,                   32, 0, stream>>>(d_tileSum, d_tileOff, d_tot);
    k_cdf       <<<BATCH * NTILES,    1024, 0, stream>>>(weight, d_max, d_tileOff, cdf);
    k_sample    <<<(BATCH * NPART) / 256, 256, 0, stream>>>(cdf, d_tot, idx, out_w);
    k_gather    <<<(BATCH * NPART * 16) / 256, 256, 0, stream>>>(state, idx, out_state);
}